// GeometricReconstructionLoss_70248485093935
// MI455X (gfx1250) — compile-verified
//
#include <hip/hip_runtime.h>
#include <hip/hip_bf16.h>

#define BATCH 16
#define NPTS  4096
#define MPTS  4096
#define ROWS_PER_WAVE 32                 // two 16-row A tiles per wave
#define ROWS_PER_WG   (8 * ROWS_PER_WAVE) // 256 rows (8 waves of 32)
#define TPB   256
#define IMAXBITS 0x7FFFFFFF              // +NaN pattern; > all finite as i32

typedef float v2f __attribute__((ext_vector_type(2)));
typedef float v8f __attribute__((ext_vector_type(8)));

// Signed-i32 min on float bit patterns: exact float order on nonnegatives,
// and any negative beats all nonnegatives -- so clamp(min_i32(...)) equals
// clamp(min_float(...)). Integer mins need no IEEE canonicalization, so the
// backend emits bare v_min_i32/v_min3_i32 (no v_max x,x quieting ops).
static __device__ __forceinline__ int imn(int a, int b) { return a < b ? a : b; }

// ---------------------------------------------------------------------------
// Prep: transform target points by T (row-vector convention, w-divide), pack
// B-side operand [-2y0,-2y1,-2y2, yy] as float4, init global col-min table,
// zero the output accumulator.
// ---------------------------------------------------------------------------
__global__ void chamfer_prep(const float* __restrict__ tar,
                             const float* __restrict__ T,
                             float4* __restrict__ BY,
                             int* __restrict__ minY,
                             float* __restrict__ out) {
    int idx = blockIdx.x * TPB + threadIdx.x;
    if (idx >= BATCH * MPTS) return;
    if (idx == 0) out[0] = 0.0f;

    int b = idx / MPTS;
    const float* t = T + b * 16;          // [4][4] row-major
    const float* p = tar + (size_t)idx * 3;
    float p0 = p[0], p1 = p[1], p2 = p[2];
    // [p,1] @ T : t_j = p0*T[0][j] + p1*T[1][j] + p2*T[2][j] + T[3][j]
    float t0 = p0 * t[0] + p1 * t[4] + p2 * t[8]  + t[12];
    float t1 = p0 * t[1] + p1 * t[5] + p2 * t[9]  + t[13];
    float t2 = p0 * t[2] + p1 * t[6] + p2 * t[10] + t[14];
    float t3 = p0 * t[3] + p1 * t[7] + p2 * t[11] + t[15];
    float inv = 1.0f / t3;
    float y0 = t0 * inv, y1 = t1 * inv, y2 = t2 * inv;
    float yy = y0 * y0 + y1 * y1 + y2 * y2;
    BY[idx] = make_float4(-2.0f * y0, -2.0f * y1, -2.0f * y2, yy);
    minY[idx] = IMAXBITS;
}

// ---------------------------------------------------------------------------
// Main: one workgroup = (batch b, 256-row chunk of X). Each wave owns 32 rows
// as two 16-row A tiles, loops over all M in 16-column tiles. Two independent
// V_WMMA_F32_16X16X4_F32 per B-tile load produce full distance tiles
// d = xx + yy - 2*x.y (biases folded into A and C operands).
// All min reductions run on signed-i32 float bit patterns; clamping to >= 0
// is deferred to the epilogue / final kernel (it commutes with the min).
// Column mins merge through ds_min_i32 -- the two lane halves cover disjoint
// row sets, so the LDS atomic itself performs the half-merge (no shuffle,
// no exec divergence in the loop).
// ---------------------------------------------------------------------------
__global__ __launch_bounds__(TPB) void chamfer_main(
        const float* __restrict__ X,
        const float4* __restrict__ BY,
        int* __restrict__ minY,
        float* __restrict__ out) {
    __shared__ int ldsMinY[MPTS];         // 16 KB of the 320 KB WGP LDS
    const int tid = threadIdx.x;
    for (int j = tid; j < MPTS; j += TPB) ldsMinY[j] = IMAXBITS;
    __syncthreads();

    const int chunksPerBatch = NPTS / ROWS_PER_WG;       // 16
    const int b     = blockIdx.x / chunksPerBatch;
    const int chunk = blockIdx.x % chunksPerBatch;
    const int wave  = tid >> 5;                          // wave32
    const int lane  = tid & 31;
    const int half  = lane >> 4;                         // A/B K split
    const int l16   = lane & 15;

    // Two A tiles: rows r0..r0+15 and r0+16..r0+31, K = [x0,x1,x2,xx].
    // ISA layout: VGPR0 = K0|K2, VGPR1 = K1|K3 across lane halves.
    const int r0 = chunk * ROWS_PER_WG + wave * ROWS_PER_WAVE + l16;
    const float* xp0 = X + ((size_t)b * NPTS + r0) * 3;
    const float* xp1 = xp0 + 16 * 3;
    float a0x = xp0[0], a0y = xp0[1], a0z = xp0[2];
    float a1x = xp1[0], a1y = xp1[1], a1z = xp1[2];
    float xx0 = a0x * a0x + a0y * a0y + a0z * a0z;
    float xx1 = a1x * a1x + a1y * a1y + a1z * a1z;
    v2f amat0, amat1;
    amat0.x = half ? a0z : a0x;  amat0.y = half ? xx0 : a0y;
    amat1.x = half ? a1z : a1x;  amat1.y = half ? xx1 : a1y;

    int accA[8], accB[8];
#pragma unroll
    for (int i = 0; i < 8; ++i) { accA[i] = IMAXBITS; accB[i] = IMAXBITS; }

    const float4* byb = BY + (size_t)b * MPTS;

    // Software-pipelined B stream: prefetch next tile (wraparound, branch-free).
    float4 q = byb[l16];
    for (int c = 0; c < MPTS; c += 16) {
        float4 qn = byb[(((c + 16) & (MPTS - 1)) + l16)];

        // B tile: col = target point, K = [-2y0,-2y1,-2y2,1]; C = yy per column.
        v2f bmat;
        bmat.x = half ? q.z : q.x;
        bmat.y = half ? 1.0f : q.y;
        v8f c0, c1;
#pragma unroll
        for (int i = 0; i < 8; ++i) { c0[i] = q.w; c1[i] = q.w; }

        // D[i][j] = xx_i + yy_j - 2 * x_i . y_j  (two independent WMMAs)
        v8f d0 = __builtin_amdgcn_wmma_f32_16x16x4_f32(
            false, amat0, false, bmat, (short)0, c0, false, false);
        v8f d1 = __builtin_amdgcn_wmma_f32_16x16x4_f32(
            false, amat1, false, bmat, (short)0, c1, false, false);

        // Row direction (min over m): elementwise i32 accumulate.
#pragma unroll
        for (int i = 0; i < 8; ++i) {
            accA[i] = imn(accA[i], __float_as_int(d0[i]));
            accB[i] = imn(accB[i], __float_as_int(d1[i]));
        }

        // Column direction: per-lane i32 min over this wave's 16 visible rows;
        // lane halves hold disjoint row sets -> ds_min_i32 merges them.
        int m = imn(__float_as_int(d0[0]), __float_as_int(d1[0]));
#pragma unroll
        for (int i = 1; i < 8; ++i)
            m = imn(m, imn(__float_as_int(d0[i]), __float_as_int(d1[i])));
        atomicMin(&ldsMinY[c + l16], m);

        q = qn;
    }

    // Finish row mins: i32 reduce across the 16 lanes of each half, then
    // decode + clamp + sum.
    float s = 0.0f;
#pragma unroll
    for (int i = 0; i < 8; ++i) {
        int rA = accA[i], rB = accB[i];
        rA = imn(rA, __shfl_xor(rA, 1, 32));
        rB = imn(rB, __shfl_xor(rB, 1, 32));
        rA = imn(rA, __shfl_xor(rA, 2, 32));
        rB = imn(rB, __shfl_xor(rB, 2, 32));
        rA = imn(rA, __shfl_xor(rA, 4, 32));
        rB = imn(rB, __shfl_xor(rB, 4, 32));
        rA = imn(rA, __shfl_xor(rA, 8, 32));
        rB = imn(rB, __shfl_xor(rB, 8, 32));
        s += fmaxf(__int_as_float(rA), 0.0f) + fmaxf(__int_as_float(rB), 0.0f);
    }
    s += __shfl_xor(s, 16, 32);          // combine the two lane-half row sets
    if (lane == 0) atomicAdd(out, s);

    // Flush this workgroup's partial column mins to the global table.
    __syncthreads();
    int* gmin = minY + (size_t)b * MPTS;
    for (int j = tid; j < MPTS; j += TPB) atomicMin(&gmin[j], ldsMinY[j]);
}

// ---------------------------------------------------------------------------
// Final: decode column mins, clamp to >= 0, wave-reduce, accumulate.
// ---------------------------------------------------------------------------
__global__ void chamfer_final(const int* __restrict__ minY,
                              float* __restrict__ out) {
    int idx = blockIdx.x * TPB + threadIdx.x;
    float v = (idx < BATCH * MPTS)
                  ? fmaxf(__int_as_float(minY[idx]), 0.0f) : 0.0f;
    v += __shfl_xor(v, 1, 32);
    v += __shfl_xor(v, 2, 32);
    v += __shfl_xor(v, 4, 32);
    v += __shfl_xor(v, 8, 32);
    v += __shfl_xor(v, 16, 32);
    if ((threadIdx.x & 31) == 0) atomicAdd(out, v);
}

extern "C" void kernel_launch(void* const* d_in, const int* in_sizes, int n_in,
                              void* d_out, int out_size, void* d_ws, size_t ws_size,
                              hipStream_t stream) {
    const float* X_v  = (const float*)d_in[0];   // [16,4096,3]
    const float* tarX = (const float*)d_in[1];   // [16,4096,3]
    const float* T    = (const float*)d_in[2];   // [16,4,4]
    float* out = (float*)d_out;                  // [1]

    // Workspace: packed B operand (float4 per target point) + i32 col-min table.
    float4* BY   = (float4*)d_ws;
    int*    minY = (int*)((char*)d_ws + (size_t)BATCH * MPTS * sizeof(float4));

    int nPrep = BATCH * MPTS;
    chamfer_prep<<<(nPrep + TPB - 1) / TPB, TPB, 0, stream>>>(tarX, T, BY, minY, out);

    int nBlocks = BATCH * (NPTS / ROWS_PER_WG);  // 256 workgroups
    chamfer_main<<<nBlocks, TPB, 0, stream>>>(X_v, BY, minY, out);

    chamfer_final<<<(nPrep + TPB - 1) / TPB, TPB, 0, stream>>>(minY, out);
}